// ADZMamba_16758962389648
// MI455X (gfx1250) — compile-verified
//
#include <hip/hip_runtime.h>
#include <cstdint>

// ---------------------------------------------------------------------------
// VMamba-style SS2D block on MI455X (gfx1250, wave32).
// GEMMs -> V_WMMA_F32_16X16X4_F32 (software-pipelined by compiler).
// Scan  -> 768 waves, 16 states/lane-group, B/C/dts streams double-buffered
//          in LDS via TENSOR_LOAD_TO_LDS (TDM) + S_WAIT_TENSORCNT.
// ---------------------------------------------------------------------------

typedef float v2f __attribute__((ext_vector_type(2)));
typedef float v8f __attribute__((ext_vector_type(8)));
typedef unsigned int u32x4 __attribute__((ext_vector_type(4)));
typedef int i32x8 __attribute__((ext_vector_type(8)));
typedef int i32x4 __attribute__((ext_vector_type(4)));

#define D_MODEL 96
#define D_INNER 192
#define NSTATE  16
#define DTRANK  6
#define KDIR    8
#define HH      64
#define WW      64
#define LL      4096
#define CHUNK   64            // scan steps staged per TDM transfer

// LDS float offsets inside k_scan's dynamic shared block
#define LB0 0
#define LB1 (CHUNK * NSTATE)                 // 1024
#define LC0 (2 * CHUNK * NSTATE)             // 2048
#define LC1 (3 * CHUNK * NSTATE)             // 3072
#define LD0 (4 * CHUNK * NSTATE)             // 4096
#define LD1 (4 * CHUNK * NSTATE + CHUNK * DTRANK)  // 4480
#define SCAN_LDS_BYTES ((4 * CHUNK * NSTATE + 2 * CHUNK * DTRANK) * 4)  // 19456

#if __has_builtin(__builtin_amdgcn_tensor_load_to_lds)
#define HAVE_TDM 1
#else
#define HAVE_TDM 0
#endif

// ---- closed-form direction permutations -----------------------------------
__device__ __forceinline__ int pos_kl(int k, int l) {
  int kk = k & 3;
  int ll = (k >= 4) ? (LL - 1 - l) : l;
  int v;
  if (kk == 0) {                       // row snake
    int r = ll >> 6, c = ll & 63;
    if (r & 1) c = 63 - c;
    v = (r << 6) | c;
  } else if (kk == 1) {                // col snake
    int c = ll >> 6, r = ll & 63;
    if (c & 1) r = 63 - r;
    v = (r << 6) | c;
  } else {                             // oblique zig-zag (65-wide transpose)
    int val;
    if (ll < 64) val = ll * 65;        // column j=0: 64 entries
    else { int t = ll - 64; int j = 1 + t / 63; int i = t - (j - 1) * 63; val = i * 65 + j; }
    int r = val >> 6, c = val & 63;
    if (kk == 3) c = 63 - c;
    v = (r << 6) | c;
  }
  return v;
}

__device__ __forceinline__ int linv_kp(int k, int p) {
  int kk = k & 3;
  int r = p >> 6, c = p & 63;
  int l;
  if (kk == 0) {
    int cc = (r & 1) ? (63 - c) : c;
    l = (r << 6) | cc;
  } else if (kk == 1) {
    int rr = (c & 1) ? (63 - r) : r;
    l = (c << 6) | rr;
  } else {
    int val = (kk == 2) ? p : ((r << 6) | (63 - c));
    int i = val / 65, j = val - i * 65;
    l = (j == 0) ? i : (64 + (j - 1) * 63 + i);
  }
  return (k >= 4) ? (LL - 1 - l) : l;
}

__device__ __forceinline__ float silu_f(float x) { return x / (1.f + __expf(-x)); }

// ---- TDM: 1-D contiguous f32 tile, global -> LDS (CDNA5 ISA) --
#if HAVE_TDM
__device__ __forceinline__ void tdm_load_1d(unsigned int lds_byte_off,
                                            const void* gptr, unsigned int nelem) {
  unsigned long long ga = (unsigned long long)(uintptr_t)gptr;
  u32x4 g0 = { 1u,                                   // count=1, user-mode, no gather
               lds_byte_off,                          // D#.lds_addr
               (unsigned int)ga,                      // global_addr[31:0]
               (unsigned int)((ga >> 32) & 0x01FFFFFFu) | (2u << 30) };  // addr[56:32] | type=2
  i32x8 g1 = { (int)(2u << 16),                       // wg_mask=0, data_size=2 (4 B)
               (int)((nelem & 0xFFFFu) << 16),        // tensor_dim0[15:0]
               (int)((nelem >> 16) | (1u << 16)),     // tensor_dim0[31:16] | tensor_dim1=1
               (int)((nelem & 0xFFFFu) << 16),        // tile_dim0 = nelem
               0,                                     // tile_dim1=0 (1-D), tile_dim2=0
               (int)nelem,                            // tensor_dim0_stride lo
               0, 0 };
  i32x4 zz = { 0, 0, 0, 0 };
#if defined(__clang_major__) && (__clang_major__ >= 23)
  i32x8 z8 = { 0, 0, 0, 0, 0, 0, 0, 0 };
  __builtin_amdgcn_tensor_load_to_lds(g0, g1, zz, zz, z8, 0);
#else
  __builtin_amdgcn_tensor_load_to_lds(g0, g1, zz, zz, 0);
#endif
}
#endif

// ---------------------------------------------------------------------------
// Kernel 1: in_proj  C[l][e] = sum_c x[c][l] * Win[e][c]   (4096x96 * 96x384)
// ---------------------------------------------------------------------------
__global__ __launch_bounds__(128) void k_in_proj(const float* __restrict__ x,
                                                 const float* __restrict__ Win,
                                                 float* __restrict__ xin_cm,
                                                 float* __restrict__ z) {
  int wave = blockIdx.x * 4 + (threadIdx.x >> 5);   // 6144 waves: 256 l-tiles x 24 e-tiles
  int lane = threadIdx.x & 31;
  int tl = wave / 24, te = wave % 24;
  int l0 = tl << 4, e0 = te << 4;
  int m = lane & 15, kh = lane >> 4;
  v8f acc = {};
  for (int c0 = 0; c0 < D_MODEL; c0 += 4) {
    v2f a, b;
    int ca = c0 + 2 * kh;
    a.x = x[(ca + 0) * LL + l0 + m];
    a.y = x[(ca + 1) * LL + l0 + m];
    b.x = Win[(e0 + m) * D_MODEL + ca + 0];
    b.y = Win[(e0 + m) * D_MODEL + ca + 1];
    acc = __builtin_amdgcn_wmma_f32_16x16x4_f32(false, a, false, b, (short)0, acc,
                                                false, false);
  }
#pragma unroll
  for (int g = 0; g < 8; ++g) {
    int l = l0 + g + 8 * kh;
    int e = e0 + m;
    float v = acc[g];
    if (e < D_INNER) xin_cm[e * LL + l] = v;
    else             z[l * D_INNER + (e - D_INNER)] = v;
  }
}

// ---------------------------------------------------------------------------
// Kernel 2: depthwise 3x3 conv (SAME) + bias + SiLU
// ---------------------------------------------------------------------------
__global__ __launch_bounds__(256) void k_dwconv(const float* __restrict__ xin,
                                                const float* __restrict__ cw,
                                                const float* __restrict__ cb,
                                                float* __restrict__ xc) {
  int idx = blockIdx.x * 256 + threadIdx.x;
  if (idx >= D_INNER * LL) return;
  int d = idx >> 12, p = idx & (LL - 1);
  int r = p >> 6, c = p & 63;
  float s = cb[d];
  const float* w = cw + d * 9;
  const float* in = xin + d * LL;
#pragma unroll
  for (int kh = 0; kh < 3; ++kh)
#pragma unroll
    for (int kw = 0; kw < 3; ++kw) {
      int rr = r + kh - 1, cc = c + kw - 1;
      if (rr >= 0 && rr < HH && cc >= 0 && cc < WW)
        s += w[kh * 3 + kw] * in[(rr << 6) + cc];
    }
  xc[idx] = silu_f(s);
}

// ---------------------------------------------------------------------------
// Kernel 3: x_proj  x_dbl[k][c][l] = sum_d Wx[k][c][d] * xc[d][pos(k,l)]
// ---------------------------------------------------------------------------
__global__ __launch_bounds__(128) void k_x_proj(const float* __restrict__ xc,
                                                const float* __restrict__ Wx,
                                                float* __restrict__ dtsraw,
                                                float* __restrict__ Bm,
                                                float* __restrict__ Cm) {
  int wave = blockIdx.x * 4 + (threadIdx.x >> 5);   // 8 * 3 * 256 = 6144 waves
  int lane = threadIdx.x & 31;
  int k = wave / (3 * 256);
  int rem = wave % (3 * 256);
  int tc = rem >> 8, tl = rem & 255;
  int c0 = tc << 4, l0 = tl << 4;
  int m = lane & 15, kh = lane >> 4;
  int pn = pos_kl(k, l0 + m);
  const float* Wk = Wx + k * 38 * D_INNER;
  int ca = c0 + m;
  int cac = (ca < 38) ? ca : 37;
  float rowvalid = (ca < 38) ? 1.f : 0.f;
  v8f acc = {};
  for (int d0 = 0; d0 < D_INNER; d0 += 4) {
    int da = d0 + 2 * kh;
    v2f a, b;
    a.x = rowvalid * Wk[cac * D_INNER + da + 0];
    a.y = rowvalid * Wk[cac * D_INNER + da + 1];
    b.x = xc[(da + 0) * LL + pn];
    b.y = xc[(da + 1) * LL + pn];
    acc = __builtin_amdgcn_wmma_f32_16x16x4_f32(false, a, false, b, (short)0, acc,
                                                false, false);
  }
#pragma unroll
  for (int g = 0; g < 8; ++g) {
    int c = c0 + g + 8 * kh;
    int l = l0 + m;
    float v = acc[g];
    if (c < 6)        dtsraw[(k * LL + l) * DTRANK + c] = v;
    else if (c < 22)  Bm[(k * LL + l) * NSTATE + (c - 6)] = v;
    else if (c < 38)  Cm[(k * LL + l) * NSTATE + (c - 22)] = v;
  }
}

// ---------------------------------------------------------------------------
// Kernel 4: selective scan with TDM double-buffering of B/C/dts streams.
// One wave per two (k,d) channels; lane = (d-half, n); shfl_xor reduction.
// ---------------------------------------------------------------------------
__global__ __launch_bounds__(32) void k_scan(const float* __restrict__ xc,
                                             const float* __restrict__ dtsraw,
                                             const float* __restrict__ Bm,
                                             const float* __restrict__ Cm,
                                             const float* __restrict__ dtw,
                                             const float* __restrict__ dtb,
                                             const float* __restrict__ Alogs,
                                             const float* __restrict__ Ds,
                                             float* __restrict__ y_all) {
  extern __shared__ float lds[];     // dynamic-only LDS -> descriptor offsets start at 0
  int blk = blockIdx.x;              // 768 = 8 dirs * 96 channel-pairs
  int k = blk / 96;
  int dp = blk % 96;
  int lane = threadIdx.x;
  int n = lane & 15;
  int d = dp * 2 + (lane >> 4);
  int kd = k * D_INNER + d;

  float Aval = -__expf(Alogs[kd * NSTATE + n]);
  float bias = dtb[kd];
  float w0 = dtw[kd * DTRANK + 0], w1 = dtw[kd * DTRANK + 1], w2 = dtw[kd * DTRANK + 2];
  float w3 = dtw[kd * DTRANK + 3], w4 = dtw[kd * DTRANK + 4], w5 = dtw[kd * DTRANK + 5];
  float Dv = Ds[kd];

  const float* dts_k = dtsraw + (size_t)k * LL * DTRANK;
  const float* B_k   = Bm + (size_t)k * LL * NSTATE;
  const float* C_k   = Cm + (size_t)k * LL * NSTATE;
  const float* u_d   = xc + (size_t)d * LL;
  float* y_k = y_all + (size_t)k * LL * D_INNER;

  const int NCH = LL / CHUNK;        // 64 chunks

  auto issue_chunk = [&](int c, int buf) {
#if HAVE_TDM
    tdm_load_1d((buf ? LB1 : LB0) * 4, B_k + (size_t)c * CHUNK * NSTATE, CHUNK * NSTATE);
    tdm_load_1d((buf ? LC1 : LC0) * 4, C_k + (size_t)c * CHUNK * NSTATE, CHUNK * NSTATE);
    tdm_load_1d((buf ? LD1 : LD0) * 4, dts_k + (size_t)c * CHUNK * DTRANK, CHUNK * DTRANK);
#else
    float* bB = lds + (buf ? LB1 : LB0);
    float* bC = lds + (buf ? LC1 : LC0);
    float* bD = lds + (buf ? LD1 : LD0);
    for (int i = lane; i < CHUNK * NSTATE; i += 32) {
      bB[i] = B_k[(size_t)c * CHUNK * NSTATE + i];
      bC[i] = C_k[(size_t)c * CHUNK * NSTATE + i];
    }
    for (int i = lane; i < CHUNK * DTRANK; i += 32)
      bD[i] = dts_k[(size_t)c * CHUNK * DTRANK + i];
#endif
  };

  issue_chunk(0, 0);

  float h = 0.f;
  for (int c = 0; c < NCH; ++c) {
    int buf = c & 1;
    const float* sB = lds + (buf ? LB1 : LB0);
    const float* sC = lds + (buf ? LC1 : LC0);
    const float* sD = lds + (buf ? LD1 : LD0);
#if HAVE_TDM
    if (c + 1 < NCH) {
      // drain pending LDS reads of the buffer the TDM is about to overwrite
      asm volatile("s_wait_dscnt 0" ::: "memory");
      issue_chunk(c + 1, buf ^ 1);
      __builtin_amdgcn_s_wait_tensorcnt(3);   // chunk c's 3 in-order ops done
    } else {
      __builtin_amdgcn_s_wait_tensorcnt(0);
    }
#endif
    for (int t = 0; t < CHUNK; ++t) {
      int l = c * CHUNK + t;
      if ((t & 15) == 0 && l + CHUNK < LL)
        __builtin_prefetch(u_d + pos_kl(k, l + CHUNK), 0, 0);  // global_prefetch on u
      int p = pos_kl(k, l);
      float u = u_d[p];
      float dt = bias + w0 * sD[t * DTRANK + 0] + w1 * sD[t * DTRANK + 1]
                      + w2 * sD[t * DTRANK + 2] + w3 * sD[t * DTRANK + 3]
                      + w4 * sD[t * DTRANK + 4] + w5 * sD[t * DTRANK + 5];
      float delta = (dt > 20.f) ? dt : log1pf(__expf(dt));      // softplus
      float dA = __expf(delta * Aval);
      float Bv = sB[t * NSTATE + n];
      h = dA * h + delta * Bv * u;
      float acc = h * sC[t * NSTATE + n];
      acc += __shfl_xor(acc, 1, 32);
      acc += __shfl_xor(acc, 2, 32);
      acc += __shfl_xor(acc, 4, 32);
      acc += __shfl_xor(acc, 8, 32);            // 16-lane group sums
      if (n == 0) y_k[l * D_INNER + d] = acc + Dv * u;
    }
#if !HAVE_TDM
    if (c + 1 < NCH) issue_chunk(c + 1, buf ^ 1);
#endif
  }
}

// ---------------------------------------------------------------------------
// Kernel 5: merge 8 directions + LayerNorm + SiLU gate
// ---------------------------------------------------------------------------
__global__ __launch_bounds__(192) void k_merge_ln(const float* __restrict__ y_all,
                                                  const float* __restrict__ z,
                                                  const float* __restrict__ ln_g,
                                                  const float* __restrict__ ln_b,
                                                  float* __restrict__ yo) {
  __shared__ float red[192];
  __shared__ float s_mu, s_rstd;
  int p = blockIdx.x;
  int d = threadIdx.x;
  float s = 0.f;
#pragma unroll
  for (int k = 0; k < KDIR; ++k) {
    int l = linv_kp(k, p);
    s += y_all[((size_t)k * LL + l) * D_INNER + d];
  }
  red[d] = s; __syncthreads();
  for (int off = 96; off >= 6; off >>= 1) {
    if (d < off) red[d] += red[d + off];
    __syncthreads();
  }
  if (d == 0) {
    float t = red[0] + red[1] + red[2] + red[3] + red[4] + red[5];
    s_mu = t / (float)D_INNER;
  }
  __syncthreads();
  float mu = s_mu;
  float dv = s - mu;
  red[d] = dv * dv; __syncthreads();
  for (int off = 96; off >= 6; off >>= 1) {
    if (d < off) red[d] += red[d + off];
    __syncthreads();
  }
  if (d == 0) {
    float t = red[0] + red[1] + red[2] + red[3] + red[4] + red[5];
    s_rstd = rsqrtf(t / (float)D_INNER + 1e-5f);
  }
  __syncthreads();
  float yn = dv * s_rstd * ln_g[d] + ln_b[d];
  float zv = z[(size_t)p * D_INNER + d];
  yo[(size_t)p * D_INNER + d] = yn * silu_f(zv);
}

// ---------------------------------------------------------------------------
// Kernel 6: out_proj  out[m][p] = sum_d Wout[m][d] * yo[p][d]
// ---------------------------------------------------------------------------
__global__ __launch_bounds__(128) void k_out_proj(const float* __restrict__ yo,
                                                  const float* __restrict__ Wout,
                                                  float* __restrict__ out) {
  int wave = blockIdx.x * 4 + (threadIdx.x >> 5);   // 6 m-tiles * 256 p-tiles
  int lane = threadIdx.x & 31;
  int tm = wave >> 8, tp = wave & 255;
  int m0 = tm << 4, p0 = tp << 4;
  int m = lane & 15, kh = lane >> 4;
  v8f acc = {};
  for (int d0 = 0; d0 < D_INNER; d0 += 4) {
    int da = d0 + 2 * kh;
    v2f a, b;
    a.x = Wout[(m0 + m) * D_INNER + da + 0];
    a.y = Wout[(m0 + m) * D_INNER + da + 1];
    b.x = yo[(p0 + m) * D_INNER + da + 0];
    b.y = yo[(p0 + m) * D_INNER + da + 1];
    acc = __builtin_amdgcn_wmma_f32_16x16x4_f32(false, a, false, b, (short)0, acc,
                                                false, false);
  }
#pragma unroll
  for (int g = 0; g < 8; ++g)
    out[(size_t)(m0 + g + 8 * kh) * LL + p0 + m] = acc[g];
}

// ---------------------------------------------------------------------------
extern "C" void kernel_launch(void* const* d_in, const int* in_sizes, int n_in,
                              void* d_out, int out_size, void* d_ws, size_t ws_size,
                              hipStream_t stream) {
  const float* x      = (const float*)d_in[0];
  const float* Win    = (const float*)d_in[1];
  const float* conv_w = (const float*)d_in[2];
  const float* conv_b = (const float*)d_in[3];
  const float* Wx     = (const float*)d_in[4];
  const float* dtw    = (const float*)d_in[5];
  const float* dtb    = (const float*)d_in[6];
  const float* Alogs  = (const float*)d_in[7];
  const float* Ds     = (const float*)d_in[8];
  const float* ln_g   = (const float*)d_in[9];
  const float* ln_b   = (const float*)d_in[10];
  const float* Wout   = (const float*)d_in[11];
  float* out = (float*)d_out;

  float* ws = (float*)d_ws;
  float* xin_cm = ws;                       // 192*4096
  float* z      = xin_cm + D_INNER * LL;    // 4096*192
  float* xc     = z      + LL * D_INNER;    // 192*4096
  float* dtsraw = xc     + D_INNER * LL;    // 8*4096*6
  float* Bm     = dtsraw + KDIR * LL * DTRANK;   // 8*4096*16
  float* Cm     = Bm     + KDIR * LL * NSTATE;   // 8*4096*16
  float* y_all  = Cm     + KDIR * LL * NSTATE;   // 8*4096*192
  float* yo     = y_all  + (size_t)KDIR * LL * D_INNER; // 4096*192

  k_in_proj <<<6144 / 4, 128, 0, stream>>>(x, Win, xin_cm, z);
  k_dwconv  <<<(D_INNER * LL + 255) / 256, 256, 0, stream>>>(xin_cm, conv_w, conv_b, xc);
  k_x_proj  <<<6144 / 4, 128, 0, stream>>>(xc, Wx, dtsraw, Bm, Cm);
  k_scan    <<<KDIR * (D_INNER / 2), 32, SCAN_LDS_BYTES, stream>>>(xc, dtsraw, Bm, Cm,
                                                                   dtw, dtb, Alogs, Ds, y_all);
  k_merge_ln<<<LL, 192, 0, stream>>>(y_all, z, ln_g, ln_b, yo);
  k_out_proj<<<1536 / 4, 128, 0, stream>>>(yo, Wout, out);
}